// ScaledGraphMAELoss_40346922778987
// MI455X (gfx1250) — compile-verified
//
#include <hip/hip_runtime.h>
#include <math.h>

#define NUM_GRAPHS 1024
#define MIN_SCALE  0.1f

typedef __attribute__((ext_vector_type(2))) float v2f;
typedef __attribute__((ext_vector_type(8))) float v8f;

__device__ __forceinline__ float readlane_f32(float v, int srclane) {
    return __int_as_float(__builtin_amdgcn_readlane(__float_as_int(v), srclane));
}

// Wave-wide f32 sum via V_WMMA_F32_16X16X4_F32.
// A (16x4 f32, 2 VGPR/lane): VGPR0 holds K=0 (lanes 0-15) / K=2 (lanes 16-31).
// Put per-lane value v in A.x, 0 in A.y. B has ones in column N=0 for all K
// (B.x = B.y = 1 on lanes 0 and 16) -> D[m][0] = v_m + v_{m+16}.
// D layout: VGPR r, lanes 0-15 -> M=r,N=lane; lanes 16-31 -> M=8+r,N=lane-16,
// so lane 0 holds D[0..7][0] and lane 16 holds D[8..15][0].
// Must be called with full EXEC (wave-uniform control flow).
__device__ __forceinline__ float wave_sum_wmma(float v) {
    const int lane = threadIdx.x & 31;
    v2f a; a.x = v; a.y = 0.0f;
    const float one = ((lane & 15) == 0) ? 1.0f : 0.0f;
    v2f b; b.x = one; b.y = one;
    v8f c = {};
    v8f d = __builtin_amdgcn_wmma_f32_16x16x4_f32(
        /*neg_a=*/false, a, /*neg_b=*/false, b,
        /*c_mod=*/(short)0, c, /*reuse_a=*/false, /*reuse_b=*/false);
    float s = d[0] + d[1] + d[2] + d[3] + d[4] + d[5] + d[6] + d[7];
    // scalar broadcasts (v_readlane_b32) instead of ds_bpermute
    return readlane_f32(s, 0) + readlane_f32(s, 16);
}

__global__ void zero_ws_kernel(float* __restrict__ ws, int n) {
    int i = blockIdx.x * blockDim.x + threadIdx.x;
    if (i < n) ws[i] = 0.0f;
}

// ws layout: [0..1023] seg_sum, [1024..2047] counts(float), [2048] total_force
__global__ void __launch_bounds__(256)
graph_mae_main_kernel(const float* __restrict__ pred,
                      const float* __restrict__ target,
                      const int*   __restrict__ batch,
                      const float* __restrict__ x,
                      float* __restrict__ ws,
                      int n, int per_wave) {
    __shared__ float blk_force;
    if (threadIdx.x == 0) blk_force = 0.0f;
    __syncthreads();

    float* seg    = ws;
    float* cnt    = ws + NUM_GRAPHS;
    float* forceg = ws + 2 * NUM_GRAPHS;

    const int lane = threadIdx.x & 31;
    const int wave = (blockIdx.x * blockDim.x + threadIdx.x) >> 5;

    int start = wave * per_wave;
    int end   = start + per_wave;
    if (start > n) start = n;
    if (end   > n) end   = n;

    float facc  = 0.0f;   // per-lane force accumulator
    float accA  = 0.0f;   // per-lane abs-sum accumulator for current run
    int   cur   = -1;     // wave-uniform (scalar) current graph id
    int   iters = 0;      // wave-uniform #iterations in current run

    const int len  = end - start;
    const int full = (len > 0) ? (len >> 5) : 0;

    for (int k = 0; k < full; ++k) {
        const int idx = start + (k << 5) + lane;
        const int p3  = idx * 3;
        const float p0 = __builtin_nontemporal_load(pred   + p3 + 0);
        const float p1 = __builtin_nontemporal_load(pred   + p3 + 1);
        const float p2 = __builtin_nontemporal_load(pred   + p3 + 2);
        const float t0 = __builtin_nontemporal_load(target + p3 + 0);
        const float t1 = __builtin_nontemporal_load(target + p3 + 1);
        const float t2 = __builtin_nontemporal_load(target + p3 + 2);
        const float a  = fabsf(p0 - t0) + fabsf(p1 - t1) + fabsf(p2 - t2);

        const int x8 = idx * 8;
        const float fx = __builtin_nontemporal_load(x + x8 + 3);
        const float fy = __builtin_nontemporal_load(x + x8 + 4);
        facc += sqrtf(fx * fx + fy * fy);

        const int bid = __builtin_nontemporal_load(batch + idx);
        const int b0  = __builtin_amdgcn_readfirstlane(bid);   // SGPR broadcast
        if (__all(bid == b0)) {          // wave-uniform branch (common case)
            if (b0 != cur) {             // scalar compare/branch
                if (cur >= 0) {          // flush previous run: one WMMA reduce
                    const float s = wave_sum_wmma(accA);
                    if (lane == 0) {
                        atomicAdd(&seg[cur], s);
                        atomicAdd(&cnt[cur], 32.0f * (float)iters);
                    }
                }
                cur = b0; accA = 0.0f; iters = 0;
            }
            accA += a;
            ++iters;
        } else {                          // graph boundary inside this warp (rare)
            if (cur >= 0) {
                const float s = wave_sum_wmma(accA);
                if (lane == 0) {
                    atomicAdd(&seg[cur], s);
                    atomicAdd(&cnt[cur], 32.0f * (float)iters);
                }
                cur = -1; accA = 0.0f; iters = 0;
            }
            atomicAdd(&seg[bid], a);
            atomicAdd(&cnt[bid], 1.0f);
        }
    }

    // Flush the last run (wave-uniform).
    if (cur >= 0) {
        const float s = wave_sum_wmma(accA);
        if (lane == 0) {
            atomicAdd(&seg[cur], s);
            atomicAdd(&cnt[cur], 32.0f * (float)iters);
        }
    }

    // Tail: fewer than 32 remaining nodes -> per-lane atomics (divergent, no WMMA).
    {
        const int idx = start + (full << 5) + lane;
        if (idx < end) {
            const int p3 = idx * 3;
            const float p0 = __builtin_nontemporal_load(pred   + p3 + 0);
            const float p1 = __builtin_nontemporal_load(pred   + p3 + 1);
            const float p2 = __builtin_nontemporal_load(pred   + p3 + 2);
            const float t0 = __builtin_nontemporal_load(target + p3 + 0);
            const float t1 = __builtin_nontemporal_load(target + p3 + 1);
            const float t2 = __builtin_nontemporal_load(target + p3 + 2);
            const float a  = fabsf(p0 - t0) + fabsf(p1 - t1) + fabsf(p2 - t2);
            const int x8 = idx * 8;
            const float fx = __builtin_nontemporal_load(x + x8 + 3);
            const float fy = __builtin_nontemporal_load(x + x8 + 4);
            facc += sqrtf(fx * fx + fy * fy);
            const int bid = __builtin_nontemporal_load(batch + idx);
            atomicAdd(&seg[bid], a);
            atomicAdd(&cnt[bid], 1.0f);
        }
    }

    // Force: WMMA wave-reduce (reconverged, full EXEC), LDS add per wave,
    // one global atomic per block.
    const float ftot = wave_sum_wmma(facc);
    if (lane == 0) atomicAdd(&blk_force, ftot);
    __syncthreads();
    if (threadIdx.x == 0) atomicAdd(forceg, blk_force);
}

__global__ void __launch_bounds__(256)
finalize_kernel(const float* __restrict__ ws, float* __restrict__ out) {
    float s = 0.0f;
    for (int g = threadIdx.x; g < NUM_GRAPHS; g += 256) {
        const float c = ws[NUM_GRAPHS + g];
        s += ws[g] / (fmaxf(c, 1.0f) * 3.0f);
    }
    for (int off = 16; off > 0; off >>= 1) s += __shfl_xor(s, off, 32);
    __shared__ float acc;
    if (threadIdx.x == 0) acc = 0.0f;
    __syncthreads();
    if ((threadIdx.x & 31) == 0) atomicAdd(&acc, s);
    __syncthreads();
    if (threadIdx.x == 0) {
        const float mean  = acc / (float)NUM_GRAPHS;
        const float force = ws[2 * NUM_GRAPHS];
        const float scale = fmaxf(force, MIN_SCALE);
        out[0] = mean * scale * 100.0f;
    }
}

extern "C" void kernel_launch(void* const* d_in, const int* in_sizes, int n_in,
                              void* d_out, int out_size, void* d_ws, size_t ws_size,
                              hipStream_t stream) {
    (void)n_in; (void)out_size; (void)ws_size;
    const float* pred   = (const float*)d_in[0];
    const float* target = (const float*)d_in[1];
    const int*   batch  = (const int*)d_in[2];
    const float* x      = (const float*)d_in[3];
    float* ws  = (float*)d_ws;
    float* out = (float*)d_out;
    const int n = in_sizes[2];   // N nodes (batch element count)

    const int nws = 2 * NUM_GRAPHS + 1;
    zero_ws_kernel<<<(nws + 255) / 256, 256, 0, stream>>>(ws, nws);

    const int blocks  = 1024;
    const int threads = 256;
    const int total_waves = blocks * threads / 32;           // 8192 waves
    const int per_wave = (n + total_waves - 1) / total_waves;
    graph_mae_main_kernel<<<blocks, threads, 0, stream>>>(pred, target, batch, x,
                                                          ws, n, per_wave);
    finalize_kernel<<<1, 256, 0, stream>>>(ws, out);
}